// Potential_Full_6347961663539
// MI455X (gfx1250) — compile-verified
//
#include <hip/hip_runtime.h>

typedef __attribute__((ext_vector_type(2))) float v2f;
typedef __attribute__((ext_vector_type(8))) float v8f;

#define PI_F     3.14159265358979323846f
#define LOG2E_F  1.44269504088896340736f
#define SIDE     128
#define PIXELS   (SIDE * SIDE)      // 16384
#define KFF      5

// ---------------------------------------------------------------------------
// Prep: one 64-byte record per (batch, atom):
//   r[0..3]  = { -2x, -2y, 1.0, x^2+y^2 }   (B-operand pairs: lanes<16 read
//              r[0..1] (K=0,1), lanes>=16 read r[2..3] (K=2,3) -- no swap)
//   r[4..8]  = w[k]  = ff_a * invb          (invb = 4*pi/ff_b)
//   r[9..13] = c[k]  = -pi*log2e*invb       (exp(z) -> exp2 argument scale)
//   r[14..15] pad (keeps records 64B / float4-aligned)
// ---------------------------------------------------------------------------
__global__ __launch_bounds__(256) void prep_rec(const float* __restrict__ coords,
                                                const float* __restrict__ ffa,
                                                const float* __restrict__ ffb,
                                                float* __restrict__ rec,
                                                int A, int nbatch) {
  int idx = blockIdx.x * blockDim.x + threadIdx.x;   // over nbatch*A
  if (idx >= nbatch * A) return;
  int a = idx % A;
  float x = coords[(size_t)idx * 3 + 0];
  float y = coords[(size_t)idx * 3 + 1];
  float* r = rec + (size_t)idx * 16;
  r[0] = -2.0f * x;
  r[1] = -2.0f * y;
  r[2] = 1.0f;
  r[3] = x * x + y * y;
#pragma unroll
  for (int k = 0; k < KFF; ++k) {
    float inv = (4.0f * PI_F) / ffb[a * KFF + k];
    r[4 + k] = ffa[a * KFF + k] * inv;
    r[9 + k] = -PI_F * LOG2E_F * inv;
  }
  r[14] = 0.0f;
  r[15] = 0.0f;
}

// ---------------------------------------------------------------------------
// Main kernel (templated on USE_TAB so the hot loop has no branches).
// One wave32 owns a 16-pixel tile (fixed i, 16 consecutive j) and 1/S of the
// atoms. Per 16-atom tile:
//   d2[16x16] = A(16x4) x B(4x16) via V_WMMA_F32_16X16X4_F32 with
//     A row (pixel m): [x, y_m, x^2+y_m^2, 1]
//     B col (atom n):  [-2x_n, -2y_n, 1, x_n^2+y_n^2]
// then 8 d2 values * 5 gaussians per lane. The k-loop keeps 8 exp results
// live at once so v_exp_f32 issues back-to-back (no TRANS v_nop hazards) and
// the 5-deep acc FMA chains are far apart.
// VGPR layouts per ISA 7.12.2: lanes 0-15 hold A/B K=0,1; lanes 16-31 K=2,3;
// C/D: lane n<16 -> (M=r, N=n), lane>=16 -> (M=r+8, N=n-16), so each lane's
// atom column is (lane & 15).
// ---------------------------------------------------------------------------
template <bool USE_TAB>
__global__ __launch_bounds__(256) void pot_kernel(
    const float* __restrict__ coords,   // [B, A, 3]
    const float* __restrict__ ffa,      // [A, 5]
    const float* __restrict__ ffb,      // [A, 5]
    const float* __restrict__ rec,      // [B, A, 16] when USE_TAB
    float* __restrict__ outbuf,         // S==1: out [B,P]; else partial [S,B,P]
    int A, int nbatch, int S) {
  const int tilesPerBatch = PIXELS / 16;                 // 1024
  const int lane = threadIdx.x & 31;
  const int wave = (int)((blockIdx.x * blockDim.x + threadIdx.x) >> 5);
  const int totalWaves = S * nbatch * tilesPerBatch;
  if (wave >= totalWaves) return;                        // wave-uniform exit

  const int s   = wave / (nbatch * tilesPerBatch);
  const int rem = wave % (nbatch * tilesPerBatch);
  const int b   = rem / tilesPerBatch;
  const int tp  = rem % tilesPerBatch;
  const int i   = tp >> 3;                               // pixel row, x = ax[i]
  const int j0  = (tp & 7) << 4;                         // 16 consecutive j
  const int half = lane >> 4;
  const int l    = lane & 15;

  const float x = (float)(i - SIDE / 2);
  const float y = (float)(j0 + l - SIDE / 2);
  v2f amat;
  if (half == 0) { amat.x = x;             amat.y = y;    }
  else           { amat.x = x * x + y * y; amat.y = 1.0f; }

  v8f acc = {};
  const float* cb = coords + (size_t)b * A * 3;
  const float* rb = rec + (size_t)b * A * 16;
  const int aPerS = A / S;
  const int t0 = s * aPerS, t1 = t0 + aPerS;

#pragma unroll 2
  for (int t = t0; t < t1; t += 16) {
    const int a = t + l;
    v2f bmat;
    float w[KFF], c[KFF];

    if (USE_TAB) {
      const float* rp = rb + (size_t)a * 16;
      bmat = *(const v2f*)(rp + (half << 1));            // pair for this half
      float4 q0 = *(const float4*)(rp + 4);              // w0..w3
      float4 q1 = *(const float4*)(rp + 8);              // w4,c0,c1,c2
      v2f    q2 = *(const v2f*)(rp + 12);                // c3,c4
      w[0] = q0.x; w[1] = q0.y; w[2] = q0.z; w[3] = q0.w;
      w[4] = q1.x; c[0] = q1.y; c[1] = q1.z; c[2] = q1.w;
      c[3] = q2.x; c[4] = q2.y;
    } else {
      float axc = cb[a * 3 + 0];
      float ayc = cb[a * 3 + 1];
      if (half == 0) { bmat.x = -2.0f * axc; bmat.y = -2.0f * ayc;           }
      else           { bmat.x = 1.0f;        bmat.y = axc * axc + ayc * ayc; }
#pragma unroll
      for (int k = 0; k < KFF; ++k) {
        float inv = (4.0f * PI_F) / ffb[a * KFF + k];
        w[k] = ffa[a * KFF + k] * inv;
        c[k] = -PI_F * LOG2E_F * inv;
      }
    }

    v8f czero = {};
    v8f d2 = __builtin_amdgcn_wmma_f32_16x16x4_f32(
        false, amat, false, bmat, (short)0, czero, false, false);

    // Keep 8 exp results live per k so the trans pipe stays full.
#pragma unroll
    for (int k = 0; k < KFF; ++k) {
      v8f e;
#pragma unroll
      for (int r = 0; r < 8; ++r)
        e[r] = __builtin_amdgcn_exp2f(c[k] * d2[r]);
#pragma unroll
      for (int r = 0; r < 8; ++r)
        acc[r] = fmaf(w[k], e[r], acc[r]);
    }
  }

  // Sum over the atom dimension: 16 lanes per half-wave hold distinct atoms.
#pragma unroll
  for (int off = 1; off < 16; off <<= 1) {
#pragma unroll
    for (int r = 0; r < 8; ++r)
      acc[r] += __shfl_xor(acc[r], off, 16);
  }

  if (l == 0) {
    float* op = outbuf + ((size_t)s * nbatch + b) * PIXELS + i * SIDE + j0 + half * 8;
#pragma unroll
    for (int r = 0; r < 8; ++r) op[r] = acc[r];
  }
}

// Deterministic reduction of the S partial grids (fixed summation order).
__global__ __launch_bounds__(256) void reduce_parts(const float* __restrict__ partial,
                                                    float* __restrict__ out,
                                                    int n, int S) {
  int idx = blockIdx.x * blockDim.x + threadIdx.x;
  if (idx >= n) return;
  float v = 0.0f;
  for (int sIdx = 0; sIdx < S; ++sIdx) v += partial[(size_t)sIdx * n + idx];
  out[idx] = v;
}

extern "C" void kernel_launch(void* const* d_in, const int* in_sizes, int n_in,
                              void* d_out, int out_size, void* d_ws, size_t ws_size,
                              hipStream_t stream) {
  const float* coords = (const float*)d_in[0];   // (B, A, 3)
  const float* ffa    = (const float*)d_in[1];   // (A, 5)
  const float* ffb    = (const float*)d_in[2];   // (A, 5)
  float* out = (float*)d_out;

  const int A = in_sizes[1] / KFF;               // 2048
  const int B = in_sizes[0] / (3 * A);           // 2
  const int n = B * PIXELS;                      // == out_size

  // Workspace: per-(batch,atom) 64B record table first, then S partial grids.
  const size_t rec_bytes = (size_t)B * A * 16 * sizeof(float);
  const int use_rec = (d_ws != nullptr && ws_size >= rec_bytes) ? 1 : 0;

  int S = 1;
  float* partial = out;
  float* rec = (float*)d_ws;
  const size_t part_off = (rec_bytes + 255) & ~(size_t)255;
  const int S_want = 8;
  if (use_rec && (A % (16 * S_want) == 0) &&
      ws_size >= part_off + (size_t)S_want * n * sizeof(float)) {
    S = S_want;                                  // 16384 waves -> fills the GPU
    partial = (float*)((char*)d_ws + part_off);
  }

  const int threads = 256;                       // 8 waves / block
  if (use_rec) {
    const int BA = B * A;
    prep_rec<<<(BA + threads - 1) / threads, threads, 0, stream>>>(
        coords, ffa, ffb, rec, A, B);
    const int totalWaves = S * B * (PIXELS / 16);
    const int blocks = (totalWaves * 32 + threads - 1) / threads;
    pot_kernel<true><<<blocks, threads, 0, stream>>>(coords, ffa, ffb, rec,
                                                     partial, A, B, S);
  } else {
    const int totalWaves = B * (PIXELS / 16);
    const int blocks = (totalWaves * 32 + threads - 1) / threads;
    pot_kernel<false><<<blocks, threads, 0, stream>>>(coords, ffa, ffb, nullptr,
                                                      out, A, B, 1);
  }

  if (S > 1)
    reduce_parts<<<(n + 255) / 256, 256, 0, stream>>>(partial, out, n, S);
}